// ScaledDotProductAttention_72584947303142
// MI455X (gfx1250) — compile-verified
//
#include <hip/hip_runtime.h>

typedef __attribute__((ext_vector_type(2))) float v2f;
typedef __attribute__((ext_vector_type(8))) float v8f;

namespace {
constexpr int Bc = 2, Hc = 16, Sc = 2048, Dc = 64;
constexpr int SROW = Sc + 2;            // padded LDS row: 8200B stride -> no bank conflicts
constexpr float INV_T = 0.125f;         // 1 / sqrt(64)
constexpr float NEGC  = -1000000000.0f; // reference NEG
}

// One block: one (b, h, 16-row query tile). 8 waves (wave32), full fp32 WMMA path.
__global__ __launch_bounds__(256, 1)
void sdpa_fwd_kernel(const float* __restrict__ q,
                     const float* __restrict__ k,
                     const float* __restrict__ v,
                     const int*   __restrict__ mask,
                     float* __restrict__ out_o,
                     float* __restrict__ out_p,
                     float* __restrict__ out_s) {
  extern __shared__ float smem[];
  float* sS = smem;              // [16][SROW] scores -> probs
  float* sO = smem + 16 * SROW;  // [16][64]   PV partial accumulators

  const int tid  = threadIdx.x;
  const int wave = tid >> 5;
  const int lane = tid & 31;
  const int m16  = lane & 15;    // M (or N) index within 16
  const int half = lane >> 4;    // which half-wave (ISA WMMA f32 layout)

  const int qt = blockIdx.x;     // 0..127
  const int h  = blockIdx.y;
  const int b  = blockIdx.z;
  const int qbase = qt * 16;

  const size_t bh = (size_t)b * Hc + h;
  const float* Q = q + (bh * Sc + qbase) * Dc;
  const float* K = k + bh * Sc * Dc;
  const float* V = v + bh * Sc * Dc;
  const int*   M = mask + ((size_t)b * Sc + qbase) * Sc;
  float* Oo = out_o + (bh * Sc + qbase) * Dc;
  float* Po = out_p + (bh * Sc + qbase) * Sc;
  float* So = out_s + (bh * Sc + qbase) * Sc;

  // ---- Q-tile A-fragments: lane holds row m16; VGPR pair = K {2*half, 2*half+1} ----
  v2f aq[16];
  {
    const float* qr = Q + m16 * Dc + 2 * half;
    #pragma unroll
    for (int kk = 0; kk < 16; ++kk) {
      aq[kk].x = qr[4 * kk];
      aq[kk].y = qr[4 * kk + 1];
    }
  }

  // ---- Phase 1: scores = (Q K^T)/T + (1-mask)*NEG ; write LDS + global attn ----
  for (int j = 0; j < 16; ++j) {
    const int kt  = wave + 8 * j;          // interleave waves over key tiles (L2 locality)
    const int kn0 = kt * 16;
    const float* kr = K + (size_t)(kn0 + m16) * Dc + 2 * half; // B frag: lane = N
    v8f acc = {};
    #pragma unroll
    for (int kk = 0; kk < 16; ++kk) {
      v2f bk;
      bk.x = kr[4 * kk];
      bk.y = kr[4 * kk + 1];
      acc = __builtin_amdgcn_wmma_f32_16x16x4_f32(
          false, aq[kk], false, bk, (short)0, acc, false, false);
    }
    const int nn = kn0 + m16;
    #pragma unroll
    for (int r = 0; r < 8; ++r) {
      const int mm = r + 8 * half;         // C/D layout: VGPR r -> M = r + 8*half
      const float am = (1.0f - (float)M[(size_t)mm * Sc + nn]) * NEGC;
      const float sc = acc[r] * INV_T + am;
      sS[mm * SROW + nn] = sc;
      __builtin_nontemporal_store(sc, So + (size_t)mm * Sc + nn);
    }
  }
  __syncthreads();

  // ---- Phase 2: row-wise softmax (each wave: 2 rows, wave32 shuffle reductions) ----
  #pragma unroll
  for (int rr = 0; rr < 2; ++rr) {
    const int row = wave * 2 + rr;
    float* srow = sS + row * SROW;
    float mx = -3.0e38f;
    for (int i = lane; i < Sc; i += 32) mx = fmaxf(mx, srow[i]);
    #pragma unroll
    for (int off = 16; off > 0; off >>= 1) mx = fmaxf(mx, __shfl_xor(mx, off, 32));
    float sum = 0.0f;
    for (int i = lane; i < Sc; i += 32) {
      const float e = __expf(srow[i] - mx);
      srow[i] = e;
      sum += e;
    }
    #pragma unroll
    for (int off = 16; off > 0; off >>= 1) sum += __shfl_xor(sum, off, 32);
    const float inv = 1.0f / sum;
    float* prow = Po + (size_t)row * Sc;
    for (int i = lane; i < Sc; i += 32) {
      const float p = srow[i] * inv;
      srow[i] = p;
      __builtin_nontemporal_store(p, prow + i);
    }
  }
  __syncthreads();

  // ---- Phase 3: output = P V  (4 n-tiles x 2 K-halves over 8 waves) ----
  const int nb    = (wave & 3) * 16;
  const int khalf = wave >> 2;
  const float* vc = V + nb + m16;          // B frag: lane = N column of V
  v8f acc = {};
  const int k0base = khalf * (Sc / 2);
  #pragma unroll 4
  for (int kk = 0; kk < Sc / 8; ++kk) {    // 256 chained WMMAs over half of K
    const int k0 = k0base + 4 * kk;
    const float* pr = sS + m16 * SROW + k0 + 2 * half;
    v2f ap; ap.x = pr[0]; ap.y = pr[1];
    v2f bv;
    bv.x = vc[(size_t)(k0 + 2 * half)     * Dc];
    bv.y = vc[(size_t)(k0 + 2 * half + 1) * Dc];
    acc = __builtin_amdgcn_wmma_f32_16x16x4_f32(
        false, ap, false, bv, (short)0, acc, false, false);
  }
  if (khalf == 1) {
    #pragma unroll
    for (int r = 0; r < 8; ++r) sO[(r + 8 * half) * Dc + nb + m16] = acc[r];
  }
  __syncthreads();
  if (khalf == 0) {
    #pragma unroll
    for (int r = 0; r < 8; ++r) {
      const int mm = r + 8 * half;
      Oo[(size_t)mm * Dc + nb + m16] = acc[r] + sO[mm * Dc + nb + m16];
    }
  }
}

extern "C" void kernel_launch(void* const* d_in, const int* in_sizes, int n_in,
                              void* d_out, int out_size, void* d_ws, size_t ws_size,
                              hipStream_t stream) {
  (void)in_sizes; (void)n_in; (void)out_size; (void)d_ws; (void)ws_size;
  const float* q    = (const float*)d_in[0];
  const float* k    = (const float*)d_in[1];
  const float* v    = (const float*)d_in[2];
  const int*   mask = (const int*)d_in[3];

  float* out_o = (float*)d_out;                         // [B,H,S,D]
  float* out_p = out_o + (size_t)Bc * Hc * Sc * Dc;     // [B,H,S,S] attn_prob
  float* out_s = out_p + (size_t)Bc * Hc * Sc * Sc;     // [B,H,S,S] attn scores

  dim3 grid(Sc / 16, Hc, Bc);
  const size_t lds = (size_t)(16 * SROW + 16 * Dc) * sizeof(float);
  sdpa_fwd_kernel<<<grid, 256, lds, stream>>>(q, k, v, mask, out_o, out_p, out_s);
}